// HeteroGNN_21285857919319
// MI455X (gfx1250) — compile-verified
//
#include <hip/hip_runtime.h>
#include <math.h>

// ---------------------------------------------------------------------------
// HeteroGCLSTM on MI455X (gfx1250).
// Because the module re-zeros h/c every layer, all SAGE convs collapse to
// their biases and the f-gate is dead (c_prev == 0).  What is left is a dense
// fused LSTM-gate pipeline, implemented with V_WMMA_F32_16X16X4_F32 (exact
// fp32) and a lane-local activation epilogue.
// ---------------------------------------------------------------------------

typedef float v2f __attribute__((ext_vector_type(2)));
typedef float v8f __attribute__((ext_vector_type(8)));

#define N_NODES 50000
#define D_HID   128
#define F_IN_   16
#define MTILES  (N_NODES / 16)   // 3125, exact

__device__ __forceinline__ float sigmoid_f(float x) {
    return 1.0f / (1.0f + __expf(-x));
}

// ---------------------------------------------------------------------------
// One LSTM layer for one node type.
// X   : [N, K] input        (K = 16 for layer 0, 128 for layer 1)
// W   : [4, 2, K, D]        gate weights (we use gates 0=i, 2=cand, 3=o)
// b   : [2, 4, 2, D]        cell biases
// cbl : [2, 4, 3, D]        SAGE lin_l biases (the only surviving conv term)
// Hout: [N, D]  relu(o * tanh(c))     Cout: [N, D]  c = sigmoid(i)*tanh(cand)
// Block = 256 threads = 8 waves; wave w owns column tile w (D = 8*16);
// blockIdx.x owns row tile (16 nodes).  EXEC is all-ones everywhere.
// ---------------------------------------------------------------------------
template <int K>
__global__ __launch_bounds__(256)
void lstm_layer_kernel(const float* __restrict__ X,
                       const float* __restrict__ W,
                       const float* __restrict__ b,
                       const float* __restrict__ cbl,
                       float* __restrict__ Hout,
                       float* __restrict__ Cout,
                       int layer, int nt) {
    const int wave  = threadIdx.x >> 5;
    const int lane  = threadIdx.x & 31;
    const int half  = lane >> 4;        // which 16-lane half
    const int lm    = lane & 15;
    const int row0  = blockIdx.x * 16;
    const int col   = wave * 16 + lm;   // output column (0..127)

    // Weight bases for the three live gates (i=0, cand=2, o=3)
    const float* Wi = W + (size_t)((0 * 2 + nt) * K) * D_HID;
    const float* Wc = W + (size_t)((2 * 2 + nt) * K) * D_HID;
    const float* Wo = W + (size_t)((3 * 2 + nt) * K) * D_HID;

    v8f acc_i = {}, acc_c = {}, acc_o = {};

    // A fragment: lane (half,lm) holds A[row0+lm][kb + 2*half + {0,1}]
    const float* arow = X + (size_t)(row0 + lm) * K + 2 * half;

#pragma unroll 4
    for (int kb = 0; kb < K; kb += 4) {
        v2f a = *(const v2f*)(arow + kb);
        const int kr = kb + 2 * half;
        v2f bi = { Wi[(size_t)kr * D_HID + col], Wi[(size_t)(kr + 1) * D_HID + col] };
        v2f bc = { Wc[(size_t)kr * D_HID + col], Wc[(size_t)(kr + 1) * D_HID + col] };
        v2f bo = { Wo[(size_t)kr * D_HID + col], Wo[(size_t)(kr + 1) * D_HID + col] };
        acc_i = __builtin_amdgcn_wmma_f32_16x16x4_f32(false, a, false, bi,
                                                      (short)0, acc_i, false, false);
        acc_c = __builtin_amdgcn_wmma_f32_16x16x4_f32(false, a, false, bc,
                                                      (short)0, acc_c, false, false);
        acc_o = __builtin_amdgcn_wmma_f32_16x16x4_f32(false, a, false, bo,
                                                      (short)0, acc_o, false, false);
    }

    // Combined bias per column: b[l,g,nt,:] + surviving SAGE lin_l biases.
    // station (nt=0): conv_gs -> cbl[l,g,1]; grid (nt=1): cbl[l,g,0]+cbl[l,g,2]
    auto gate_bias = [&](int g) -> float {
        float v = b[(size_t)((layer * 4 + g) * 2 + nt) * D_HID + col];
        if (nt == 0) {
            v += cbl[(size_t)((layer * 4 + g) * 3 + 1) * D_HID + col];
        } else {
            v += cbl[(size_t)((layer * 4 + g) * 3 + 0) * D_HID + col]
               + cbl[(size_t)((layer * 4 + g) * 3 + 2) * D_HID + col];
        }
        return v;
    };
    const float bgi = gate_bias(0);
    const float bgc = gate_bias(2);
    const float bgo = gate_bias(3);

    // C/D layout: acc[v] is element (row = row0 + v + 8*half, col)
#pragma unroll
    for (int v = 0; v < 8; ++v) {
        const int m  = row0 + v + 8 * half;
        const float iv = sigmoid_f(acc_i[v] + bgi);
        const float tv = tanhf(acc_c[v] + bgc);
        const float ov = sigmoid_f(acc_o[v] + bgo);
        const float cv = iv * tv;                       // c_prev == 0
        const float hv = fmaxf(ov * tanhf(cv), 0.0f);   // relu(h)
        Cout[(size_t)m * D_HID + col] = cv;
        Hout[(size_t)m * D_HID + col] = hv;
    }
}

// ---------------------------------------------------------------------------
// Final linear: fin[nt] = h1relu[nt] @ linW[nt] + linb[nt]   ([N,128]@[128,16])
// One wave per 16-node tile; single 16x16 output tile (F_IN = 16).
// ---------------------------------------------------------------------------
__global__ __launch_bounds__(256)
void final_lin_kernel(const float* __restrict__ H1,    // [2, N, D]
                      const float* __restrict__ linW,  // [2, D, F_IN]
                      const float* __restrict__ linb,  // [2, F_IN]
                      float* __restrict__ Fout) {      // [2, N, F_IN]
    const int nt    = blockIdx.y;
    const int wave  = threadIdx.x >> 5;
    const int lane  = threadIdx.x & 31;
    const int half  = lane >> 4;
    const int lm    = lane & 15;
    const int mtile = blockIdx.x * 8 + wave;
    if (mtile >= MTILES) return;   // wave-uniform; surviving waves keep EXEC=all-1

    const float* A  = H1 + (size_t)nt * N_NODES * D_HID;
    const float* Bw = linW + (size_t)nt * D_HID * F_IN_;

    v8f acc = {};
    const float* arow = A + (size_t)(mtile * 16 + lm) * D_HID + 2 * half;

#pragma unroll 4
    for (int kb = 0; kb < D_HID; kb += 4) {
        v2f a = *(const v2f*)(arow + kb);
        const int kr = kb + 2 * half;
        v2f bv = { Bw[kr * F_IN_ + lm], Bw[(kr + 1) * F_IN_ + lm] };
        acc = __builtin_amdgcn_wmma_f32_16x16x4_f32(false, a, false, bv,
                                                    (short)0, acc, false, false);
    }

    const float bias = linb[nt * F_IN_ + lm];
    float* out = Fout + (size_t)nt * N_NODES * F_IN_;
#pragma unroll
    for (int v = 0; v < 8; ++v) {
        const int m = mtile * 16 + v + 8 * half;
        out[(size_t)m * F_IN_ + lm] = acc[v] + bias;
    }
}

// ---------------------------------------------------------------------------
// Launcher.  Output layout (flat, fp32):
//   [0]        h0_station [N,128]     [1] h0_grid
//   [2]        h1_station              [3] h1_grid
//   [4]        fin_station [N,16]      [5] fin_grid
//   [6..9]     c0_st, c0_gr, c1_st, c1_gr  [N,128] each
// Edge indices (d_in[2..4]) and convWl/convWr (d_in[8], d_in[10]) are
// mathematically unused because h is re-zeroed each layer.
// ---------------------------------------------------------------------------
extern "C" void kernel_launch(void* const* d_in, const int* in_sizes, int n_in,
                              void* d_out, int out_size, void* d_ws, size_t ws_size,
                              hipStream_t stream) {
    const float* x_st  = (const float*)d_in[0];
    const float* x_gr  = (const float*)d_in[1];
    const float* W0    = (const float*)d_in[5];   // (4,2,16,128)
    const float* W1    = (const float*)d_in[6];   // (4,2,128,128)
    const float* b     = (const float*)d_in[7];   // (2,4,2,128)
    const float* cbl   = (const float*)d_in[9];   // (2,4,3,128)
    const float* linW  = (const float*)d_in[11];  // (2,128,16)
    const float* linb  = (const float*)d_in[12];  // (2,16)

    float* out = (float*)d_out;
    const size_t ND = (size_t)N_NODES * D_HID;
    const size_t NF = (size_t)N_NODES * F_IN_;
    float* h0  = out;                    // + nt*ND
    float* h1  = out + 2 * ND;
    float* fin = out + 4 * ND;
    float* c0  = out + 4 * ND + 2 * NF;
    float* c1  = out + 6 * ND + 2 * NF;

    dim3 blk(256);
    dim3 grd(MTILES);

    // Layer 0: [N,16] @ [16,128] x 3 gates, fused LSTM epilogue
    lstm_layer_kernel<F_IN_><<<grd, blk, 0, stream>>>(x_st, W0, b, cbl, h0,      c0,      0, 0);
    lstm_layer_kernel<F_IN_><<<grd, blk, 0, stream>>>(x_gr, W0, b, cbl, h0 + ND, c0 + ND, 0, 1);

    // Layer 1: [N,128] @ [128,128] x 3 gates (reads h0 straight from d_out)
    lstm_layer_kernel<D_HID><<<grd, blk, 0, stream>>>(h0,      W1, b, cbl, h1,      c1,      1, 0);
    lstm_layer_kernel<D_HID><<<grd, blk, 0, stream>>>(h0 + ND, W1, b, cbl, h1 + ND, c1 + ND, 1, 1);

    // Final projection: [N,128] @ [128,16]
    final_lin_kernel<<<dim3((MTILES + 7) / 8, 2), blk, 0, stream>>>(h1, linW, linb, fin);
}